// FloatQuantizer_52742198395452
// MI455X (gfx1250) — compile-verified
//
#include <hip/hip_runtime.h>

// Elementwise e4m3 (bias-7, min-normal 2^-6, subnormal LSB 2^-9, saturate 240)
// fake-quantize:  out = q(x / s) * s,  s = scale[0].
// Memory-bound streaming op: 512 MiB in + 512 MiB out @ 23.3 TB/s -> ~46 us.
// The quantizer grid is exactly OCP E4M3, so after pre-clamping to +-240
// (grid-equivalent to round-then-saturate at 240) the whole quantize is two
// hardware conversions: v_cvt_pk_fp8_f32 (RNE) + v_cvt_pk_f32_fp8.
// FP8 cvt builtins are used UNCONDITIONALLY this round so the compile result
// tells us definitively whether gfx1250 selects them.

typedef float f32x4 __attribute__((ext_vector_type(4)));
typedef float f32x2 __attribute__((ext_vector_type(2)));

__device__ __forceinline__ float clamp240(float y) {
    return fminf(fmaxf(y, -240.0f), 240.0f);   // expect v_med3_f32 / min+max
}

// Quantize 4 lanes through the FP8 conversion pipe.
__device__ __forceinline__ f32x4 quant4(f32x4 y) {
    float cx = clamp240(y.x);
    float cy = clamp240(y.y);
    float cz = clamp240(y.z);
    float cw = clamp240(y.w);
    int lo = __builtin_amdgcn_cvt_pk_fp8_f32(cx, cy, 0, false);
    int hi = __builtin_amdgcn_cvt_pk_fp8_f32(cz, cw, 0, false);
    f32x2 ab = __builtin_amdgcn_cvt_pk_f32_fp8(lo, false);
    f32x2 cd = __builtin_amdgcn_cvt_pk_f32_fp8(hi, false);
    f32x4 o;
    o.x = ab.x; o.y = ab.y; o.z = cd.x; o.w = cd.y;
    return o;
}

__device__ __forceinline__ float quant1(float y) {
    int p = __builtin_amdgcn_cvt_pk_fp8_f32(clamp240(y), 0.0f, 0, false);
    f32x2 r = __builtin_amdgcn_cvt_pk_f32_fp8(p, false);
    return r.x;
}

__global__ __launch_bounds__(256) void fq4_kernel(const float* __restrict__ x,
                                                  const float* __restrict__ scale,
                                                  float* __restrict__ out,
                                                  long long n4) {
    const float s   = scale[0];
    const float inv = 1.0f / s;                 // uniform; hoisted reciprocal
    const f32x4* __restrict__ x4 = (const f32x4*)x;
    f32x4* __restrict__ o4       = (f32x4*)out;

    const long long stride = (long long)gridDim.x * blockDim.x;
    const long long step2  = stride << 1;
    long long i = (long long)blockIdx.x * blockDim.x + threadIdx.x;

    // Pairwise grid-stride: issue two NT b128 loads back-to-back per iteration
    // (2x outstanding loads per wave) before any dependent compute/store.
    for (; i + stride < n4; i += step2) {
        __builtin_prefetch((const void*)(x4 + i + step2), 0, 1);  // global_prefetch_b8
        f32x4 a = __builtin_nontemporal_load(x4 + i);
        f32x4 b = __builtin_nontemporal_load(x4 + i + stride);
        f32x4 qa = quant4(a * inv) * s;
        f32x4 qb = quant4(b * inv) * s;
        __builtin_nontemporal_store(qa, o4 + i);
        __builtin_nontemporal_store(qb, o4 + i + stride);
    }
    if (i < n4) {
        f32x4 a  = __builtin_nontemporal_load(x4 + i);
        f32x4 qa = quant4(a * inv) * s;
        __builtin_nontemporal_store(qa, o4 + i);
    }
}

__global__ __launch_bounds__(256) void fq1_kernel(const float* __restrict__ x,
                                                  const float* __restrict__ scale,
                                                  float* __restrict__ out,
                                                  long long n) {
    const float s   = scale[0];
    const float inv = 1.0f / s;
    long long i = (long long)blockIdx.x * blockDim.x + threadIdx.x;
    if (i < n) {
        out[i] = quant1(x[i] * inv) * s;
    }
}

extern "C" void kernel_launch(void* const* d_in, const int* in_sizes, int n_in,
                              void* d_out, int out_size, void* d_ws, size_t ws_size,
                              hipStream_t stream) {
    (void)n_in; (void)d_ws; (void)ws_size;
    const float* x     = (const float*)d_in[0];
    const float* scale = (const float*)d_in[1];
    float*       out   = (float*)d_out;

    const long long n  = in_sizes[0] > 0 ? (long long)in_sizes[0]
                                         : (long long)out_size;
    const long long n4 = n >> 2;           // base pointers are 16B-aligned

    if (n4 > 0) {
        const int threads = 256;
        long long want = (n4 + threads - 1) / threads;
        int blocks = (int)(want > 65536 ? 65536 : want);
        if (blocks < 1) blocks = 1;
        fq4_kernel<<<blocks, threads, 0, stream>>>(x, scale, out, n4);
    }
    const long long rem = n - (n4 << 2);
    if (rem > 0) {
        fq1_kernel<<<1, 256, 0, stream>>>(x + (n4 << 2), scale, out + (n4 << 2), rem);
    }
}